// YOLOLoss_46514495816330
// MI455X (gfx1250) — compile-verified
//
#include <hip/hip_runtime.h>

typedef __attribute__((ext_vector_type(2))) float v2f;
typedef __attribute__((ext_vector_type(8))) float v8f;

#define NCLS 80
#define NA   2
#define NB   16
#define NT   32

// Workspace layout (floats):
//  [0    .. 1049]  obj softplus block partials (800 + 200 + 50)
//  [1050 .. 1113]  obj mask-subtraction partials (negated), 64 slots
//  [1120 .. 1167]  box block partials (48)
//  [1184 .. 2143]  cls block partials (960)
#define WS_OBJ0 0
#define WS_OBJ1 800
#define WS_OBJ2 1000
#define WS_MASK 1050
#define WS_BOX  1120
#define WS_CLS  1184

__device__ __forceinline__ float softplusf(float x) {
  // stable: max(x,0) + log1p(exp(-|x|))
  return fmaxf(x, 0.0f) + log1pf(expf(-fabsf(x)));
}

__device__ __forceinline__ float block_reduce_256(float v) {
  __shared__ float s[256];
  const int t = threadIdx.x;
  s[t] = v;
  __syncthreads();
#pragma unroll
  for (int off = 128; off > 0; off >>= 1) {
    if (t < off) s[t] += s[t + off];
    __syncthreads();
  }
  return s[0];
}

// -------- obj: softplus over all objectness logits (channel 4, stride 85) ----
__global__ void __launch_bounds__(256)
obj_softplus_kernel(const float* __restrict__ pred, float* __restrict__ partials) {
  const size_t idx = (size_t)blockIdx.x * 256 + threadIdx.x;
  const float x = pred[idx * 85 + 4];
  const float s = block_reduce_256(softplusf(x));
  if (threadIdx.x == 0) partials[blockIdx.x] = s;
}

// -------- obj mask: -sum of pred_obj at unique (b,cell) cells, all anchors ---
__global__ void obj_mask_kernel(const float* __restrict__ p0,
                                const float* __restrict__ p1,
                                const float* __restrict__ p2,
                                const float* __restrict__ tg,
                                float* __restrict__ partials) {
  const int tid = threadIdx.x;   // 64 threads; 48 active
  float s = 0.0f;
  if (tid < 3 * NB) {
    const int level = tid / NB;
    const int b = tid % NB;
    const float* pred = (level == 0) ? p0 : ((level == 1) ? p1 : p2);
    const int G = (level == 0) ? 80 : ((level == 1) ? 40 : 20);
    int cells[NT];
    for (int t = 0; t < NT; ++t) {
      const float gx = tg[(b * NT + t) * 5 + 1] * (float)G;
      const float gy = tg[(b * NT + t) * 5 + 2] * (float)G;
      const int gi = min(max((int)gx, 0), G - 1);
      const int gj = min(max((int)gy, 0), G - 1);
      const int cell = gj * G + gi;
      cells[t] = cell;
      bool dup = false;
      for (int u = 0; u < t; ++u)
        if (cells[u] == cell) { dup = true; break; }
      if (!dup) {
        for (int a = 0; a < NA; ++a) {
          const size_t base = ((size_t)(b * NA + a) * G * G + cell) * 85 + 4;
          s += pred[base];
        }
      }
    }
  }
  partials[tid] = -s;  // negated so obj total = sum(softplus) + sum(mask part)
}

// -------- box: sum of squared diff at gathered cells (4 channels) -----------
__global__ void __launch_bounds__(256)
box_kernel(const float* __restrict__ p0, const float* __restrict__ p1,
           const float* __restrict__ p2, const float* __restrict__ tg,
           float* __restrict__ partials) {
  const int g = blockIdx.x * 256 + threadIdx.x;    // 12288 total
  const int c = g & 3;
  int item = g >> 2;                               // (level,b,a,t)
  const int t = item & 31; item >>= 5;
  const int a = item & 1;  item >>= 1;
  const int b = item & 15; item >>= 4;
  const int level = item;
  const float* pred = (level == 0) ? p0 : ((level == 1) ? p1 : p2);
  const int G = (level == 0) ? 80 : ((level == 1) ? 40 : 20);
  const float gx = tg[(b * NT + t) * 5 + 1] * (float)G;
  const float gy = tg[(b * NT + t) * 5 + 2] * (float)G;
  const int gi = min(max((int)gx, 0), G - 1);
  const int gj = min(max((int)gy, 0), G - 1);
  const size_t base = ((size_t)(b * NA + a) * G * G + (size_t)gj * G + gi) * 85;
  const float pv = pred[base + c];
  const float tv = tg[(b * NT + t) * 5 + 1 + c];
  const float d = pv - tv;
  const float s = block_reduce_256(d * d);
  if (threadIdx.x == 0) partials[blockIdx.x] = s;
}

// -------- cls: BCE-with-logits vs one-hot at gathered cells (80 channels) ---
__global__ void __launch_bounds__(256)
cls_kernel(const float* __restrict__ p0, const float* __restrict__ p1,
           const float* __restrict__ p2, const float* __restrict__ tg,
           float* __restrict__ partials) {
  const int g = blockIdx.x * 256 + threadIdx.x;    // 245760 total
  const int c = g % NCLS;
  int item = g / NCLS;
  const int t = item & 31; item >>= 5;
  const int a = item & 1;  item >>= 1;
  const int b = item & 15; item >>= 4;
  const int level = item;
  const float* pred = (level == 0) ? p0 : ((level == 1) ? p1 : p2);
  const int G = (level == 0) ? 80 : ((level == 1) ? 40 : 20);
  const float gx = tg[(b * NT + t) * 5 + 1] * (float)G;
  const float gy = tg[(b * NT + t) * 5 + 2] * (float)G;
  const int gi = min(max((int)gx, 0), G - 1);
  const int gj = min(max((int)gy, 0), G - 1);
  const size_t base = ((size_t)(b * NA + a) * G * G + (size_t)gj * G + gi) * 85;
  const float x = pred[base + 5 + c];
  const int tcls = (int)tg[(b * NT + t) * 5 + 0];
  const float z = (c == tcls) ? 1.0f : 0.0f;
  const float v = softplusf(x) - x * z;
  const float s = block_reduce_256(v);
  if (threadIdx.x == 0) partials[blockIdx.x] = s;
}

// -------- final: single-wave WMMA reduction (ones-matrix trick, full f32) ---
// D = ones(16x4) x B(4x16) + C  =>  D[m][n] = sum_k B[k][n] + C[m][n].
// Each issue folds 64 partials into the accumulator; all D rows identical, so
// sum of c[0] over 32 lanes = 2 * total.
__device__ __forceinline__ float wave_wmma_reduce(const float* __restrict__ data, int n) {
  v2f a; a.x = 1.0f; a.y = 1.0f;
  v8f c = {};
  const int lane = threadIdx.x;  // 0..31, EXEC all ones (uniform control flow)
  for (int base = 0; base < n; base += 64) {
    const int i0 = base + lane * 2;
    v2f b;
    b.x = (i0 < n)     ? data[i0]     : 0.0f;
    b.y = (i0 + 1 < n) ? data[i0 + 1] : 0.0f;
    c = __builtin_amdgcn_wmma_f32_16x16x4_f32(false, a, false, b, (short)0, c,
                                              false, false);
  }
  float v = c[0];
#pragma unroll
  for (int off = 16; off >= 1; off >>= 1)
    v += __shfl_xor(v, off);
  return v * 0.5f;
}

__global__ void final_kernel(const float* __restrict__ ws, float* __restrict__ out) {
  float obj = wave_wmma_reduce(ws + WS_OBJ0, 1114);  // 1050 obj + 64 mask slots
  float box = wave_wmma_reduce(ws + WS_BOX, 48);
  float cls = wave_wmma_reduce(ws + WS_CLS, 960);
  const float nt = 3.0f * (float)(NB * NA * NT);     // 3072
  box /= nt; obj /= nt; cls /= nt;
  if (threadIdx.x == 0) {
    out[0] = box + obj + cls;
    out[1] = box;
    out[2] = obj;
    out[3] = cls;
  }
}

extern "C" void kernel_launch(void* const* d_in, const int* in_sizes, int n_in,
                              void* d_out, int out_size, void* d_ws, size_t ws_size,
                              hipStream_t stream) {
  const float* p0 = (const float*)d_in[0];
  const float* p1 = (const float*)d_in[1];
  const float* p2 = (const float*)d_in[2];
  const float* tg = (const float*)d_in[3];
  float* out = (float*)d_out;
  float* ws  = (float*)d_ws;

  // obj softplus: N/256 blocks per level (204800, 51200, 12800 elements)
  obj_softplus_kernel<<<800, 256, 0, stream>>>(p0, ws + WS_OBJ0);
  obj_softplus_kernel<<<200, 256, 0, stream>>>(p1, ws + WS_OBJ1);
  obj_softplus_kernel<<<50,  256, 0, stream>>>(p2, ws + WS_OBJ2);
  obj_mask_kernel<<<1, 64, 0, stream>>>(p0, p1, p2, tg, ws + WS_MASK);
  box_kernel<<<48,  256, 0, stream>>>(p0, p1, p2, tg, ws + WS_BOX);
  cls_kernel<<<960, 256, 0, stream>>>(p0, p1, p2, tg, ws + WS_CLS);
  final_kernel<<<1, 32, 0, stream>>>(ws, out);
}